// GATNet_18743237280528
// MI455X (gfx1250) — compile-verified
//
#include <hip/hip_runtime.h>
#include <cstdint>
#include <cstddef>

#define N_NODES  50000
#define N_EDGES  1600000
#define N_GRAPHS 64

typedef float v2f __attribute__((ext_vector_type(2)));
typedef float v8f __attribute__((ext_vector_type(8)));

// ---------------------------------------------------------------------------
// GEMM: O[N, NC] = A[N,128] @ W[128, NC]   (NC = 128 or 64)
// One block per 16-row tile; one wave32 per 16-col tile (blockDim = NC*2).
// A tile staged in LDS (stride 132 to avoid bank conflicts), WMMA f32 16x16x4.
// ---------------------------------------------------------------------------
template<int NC>
__global__ __launch_bounds__(NC * 2)
void gemm_wmma(const float* __restrict__ A, const float* __restrict__ W,
               float* __restrict__ O) {
  constexpr int K = 128;
  __shared__ float As[16 * 132];

  const int rowTile = blockIdx.x;
  const float* Abase = A + (size_t)rowTile * 16 * K;  // 16 contiguous rows
  for (int i = threadIdx.x; i < 16 * K; i += NC * 2)
    As[(i >> 7) * 132 + (i & 127)] = Abase[i];
  __syncthreads();

  const int lane    = threadIdx.x & 31;
  const int colTile = threadIdx.x >> 5;          // 0 .. NC/16-1
  const int r16     = lane & 15;
  const int khalf   = (lane >> 4) << 1;          // 0 or 2
  const int col     = colTile * 16 + r16;
  const float* Wc   = W + col;

  v8f acc = {};
#pragma unroll
  for (int k0 = 0; k0 < K; k0 += 4) {
    // A fragment: lane(0-15)->row r16 K={k0,k0+1}; lane(16-31)->K={k0+2,k0+3}
    const float2 av = *(const float2*)&As[r16 * 132 + k0 + khalf];
    v2f a; a.x = av.x; a.y = av.y;
    // B fragment: same half-split over K, column = lane&15
    v2f b;
    b.x = Wc[(size_t)(k0 + khalf)     * NC];
    b.y = Wc[(size_t)(k0 + khalf + 1) * NC];
    acc = __builtin_amdgcn_wmma_f32_16x16x4_f32(false, a, false, b,
                                                (short)0, acc, false, false);
  }

  // C/D layout: VGPR v, lane -> row = v + 8*(lane>>4), col = lane&15
  const int orow = rowTile * 16 + ((lane >> 4) << 3);
  float* Ob = O + (size_t)orow * NC + col;
#pragma unroll
  for (int v = 0; v < 8; v++)
    Ob[(size_t)v * NC] = acc[v];
}

// ---------------------------------------------------------------------------
// Degree / norm
// ---------------------------------------------------------------------------
__global__ void deg_init(float* deg) {
  int i = blockIdx.x * blockDim.x + threadIdx.x;
  if (i < N_NODES) deg[i] = 1.0f;  // self-loop weight
}

__global__ void deg_accum(const int* __restrict__ dst,
                          const float* __restrict__ ew, float* deg) {
  int e = blockIdx.x * blockDim.x + threadIdx.x;
  if (e < N_EDGES) atomicAdd(&deg[dst[e]], ew[e]);
}

__global__ void dinv_k(const float* __restrict__ deg, float* dinv) {
  int i = blockIdx.x * blockDim.x + threadIdx.x;
  if (i < N_NODES) {
    float d = deg[i];
    dinv[i] = d > 0.f ? rsqrtf(d) : 0.f;
  }
}

// ---------------------------------------------------------------------------
// agg[i,:] = dinv[i]^2 * xw[i,:]   (self-loop term; also zero-initializes agg)
// ---------------------------------------------------------------------------
template<int C>
__global__ void agg_init(const float* __restrict__ xw,
                         const float* __restrict__ dinv, float* __restrict__ agg) {
  size_t i = (size_t)blockIdx.x * blockDim.x + threadIdx.x;
  if (i >= (size_t)N_NODES * C) return;
  float di = dinv[i / C];
  agg[i] = di * di * xw[i];
}

// ---------------------------------------------------------------------------
// Edge scatter: one wave32 per edge, VEC consecutive channels per lane.
// agg[dst] += (dinv[src]*ew*dinv[dst]) * xw[src]   via global f32 atomics (L2).
// ---------------------------------------------------------------------------
template<int VEC>
__global__ void edge_scatter(const float* __restrict__ xw,
                             const int* __restrict__ src,
                             const int* __restrict__ dst,
                             const float* __restrict__ ew,
                             const float* __restrict__ dinv,
                             float* __restrict__ agg) {
  const int wid  = (int)(((size_t)blockIdx.x * blockDim.x + threadIdx.x) >> 5);
  const int lane = threadIdx.x & 31;
  if (wid >= N_EDGES) return;
  const int s = src[wid], d = dst[wid];
  const float coef = dinv[s] * ew[wid] * dinv[d];
  constexpr int C = 32 * VEC;
  const float* xs = xw + (size_t)s * C + lane * VEC;
  float* ad       = agg + (size_t)d * C + lane * VEC;
#pragma unroll
  for (int v = 0; v < VEC; v++) atomicAdd(&ad[v], coef * xs[v]);
}

template<int C>
__global__ void relu_bias(float* __restrict__ h, const float* __restrict__ b) {
  size_t i = (size_t)blockIdx.x * blockDim.x + threadIdx.x;
  if (i >= (size_t)N_NODES * C) return;
  h[i] = fmaxf(h[i] + b[i & (C - 1)], 0.f);
}

// ---------------------------------------------------------------------------
// Gate MLP per node (one wave32 per node) + segment-max via uint-key atomicMax
// ---------------------------------------------------------------------------
__global__ void zero_small(unsigned* gmax_key, float* den, float* x2) {
  for (int i = threadIdx.x; i < N_GRAPHS; i += blockDim.x) {
    gmax_key[i] = 0u;          // encodes "most negative"
    den[i] = 0.f;
  }
  for (int i = threadIdx.x; i < N_GRAPHS * 64; i += blockDim.x) x2[i] = 0.f;
}

__global__ void gate_kernel(const float* __restrict__ x1,
                            const int* __restrict__ batch,
                            const float* __restrict__ gW1,
                            const float* __restrict__ gb1,
                            const float* __restrict__ gW2,
                            const float* __restrict__ gb2,
                            float* __restrict__ g, unsigned* gmax_key) {
  const int n    = (int)(((size_t)blockIdx.x * blockDim.x + threadIdx.x) >> 5);
  const int lane = threadIdx.x & 31;
  if (n >= N_NODES) return;
  const float* xr = x1 + (size_t)n * 64;
  float acc = gb1[lane];
#pragma unroll 8
  for (int k = 0; k < 64; k++) acc += xr[k] * gW1[k * 32 + lane];
  acc = fmaxf(acc, 0.f) * gW2[lane];
#pragma unroll
  for (int off = 16; off > 0; off >>= 1) acc += __shfl_xor(acc, off, 32);
  if (lane == 0) {
    float gv = acc + gb2[0];
    g[n] = gv;
    unsigned u = __float_as_uint(gv);
    unsigned key = (u & 0x80000000u) ? ~u : (u | 0x80000000u);  // monotonic
    atomicMax(&gmax_key[batch[n]], key);
  }
}

__global__ void softmax_num(const float* __restrict__ g,
                            const int* __restrict__ batch,
                            const unsigned* __restrict__ gmax_key,
                            float* __restrict__ ex, float* __restrict__ den) {
  int n = blockIdx.x * blockDim.x + threadIdx.x;
  if (n >= N_NODES) return;
  int b = batch[n];
  unsigned key = gmax_key[b];
  float gm = (key & 0x80000000u) ? __uint_as_float(key ^ 0x80000000u)
                                 : __uint_as_float(~key);
  float e = expf(g[n] - gm);
  ex[n] = e;
  atomicAdd(&den[b], e);
}

__global__ void attn_pool(const float* __restrict__ x1,
                          const int* __restrict__ batch,
                          const float* __restrict__ ex,
                          const float* __restrict__ den, float* __restrict__ x2) {
  const int n    = (int)(((size_t)blockIdx.x * blockDim.x + threadIdx.x) >> 5);
  const int lane = threadIdx.x & 31;
  if (n >= N_NODES) return;
  const int b = batch[n];
  const float alpha = ex[n] / den[b];
  const float* xr = x1 + (size_t)n * 64 + lane * 2;
  float* o = x2 + b * 64 + lane * 2;
  atomicAdd(&o[0], alpha * xr[0]);
  atomicAdd(&o[1], alpha * xr[1]);
}

// ---------------------------------------------------------------------------
// Head: relu(x2@fcW1 + b) @ fcW2 + b  ->  log_softmax, one block per graph
// ---------------------------------------------------------------------------
__global__ void final_fc(const float* __restrict__ x2,
                         const float* __restrict__ fcW1, const float* __restrict__ fcb1,
                         const float* __restrict__ fcW2, const float* __restrict__ fcb2,
                         float* __restrict__ out) {
  __shared__ float h2[128];
  __shared__ float lg[2];
  const int g = blockIdx.x;
  const int j = threadIdx.x;  // 0..127
  const float* xg = x2 + g * 64;
  float acc = fcb1[j];
#pragma unroll 8
  for (int k = 0; k < 64; k++) acc += xg[k] * fcW1[k * 128 + j];
  h2[j] = fmaxf(acc, 0.f);
  __syncthreads();
  if (j < 2) {
    float l = fcb2[j];
    for (int k = 0; k < 128; k++) l += h2[k] * fcW2[k * 2 + j];
    lg[j] = l;
  }
  __syncthreads();
  if (j == 0) {
    float m = fmaxf(lg[0], lg[1]);
    float lse = m + logf(expf(lg[0] - m) + expf(lg[1] - m));
    out[g * 2 + 0] = lg[0] - lse;
    out[g * 2 + 1] = lg[1] - lse;
  }
}

// ---------------------------------------------------------------------------
extern "C" void kernel_launch(void* const* d_in, const int* in_sizes, int n_in,
                              void* d_out, int out_size, void* d_ws, size_t ws_size,
                              hipStream_t stream) {
  const float* x     = (const float*)d_in[0];
  const int*   eidx  = (const int*)  d_in[1];
  const float* eattr = (const float*)d_in[2];
  const int*   batch = (const int*)  d_in[3];
  const float* W1  = (const float*)d_in[4];
  const float* b1  = (const float*)d_in[5];
  const float* W2  = (const float*)d_in[6];
  const float* b2  = (const float*)d_in[7];
  const float* gW1 = (const float*)d_in[8];
  const float* gb1 = (const float*)d_in[9];
  const float* gW2 = (const float*)d_in[10];
  const float* gb2 = (const float*)d_in[11];
  const float* fcW1 = (const float*)d_in[12];
  const float* fcb1 = (const float*)d_in[13];
  const float* fcW2 = (const float*)d_in[14];
  const float* fcb2 = (const float*)d_in[15];
  float* out = (float*)d_out;

  const int* src = eidx;
  const int* dst = eidx + N_EDGES;

  // workspace layout (bytes)
  char* ws = (char*)d_ws;
  float*    bufA     = (float*)(ws);                         // 50000*128 (xw1 -> xw2)
  float*    bufB     = (float*)(ws + 25600000);              // 50000*128 (agg1/h -> agg2/x1)
  float*    deg      = (float*)(ws + 51200000);              // 50000
  float*    dinv     = (float*)(ws + 51400000);              // 50000
  float*    gbuf     = (float*)(ws + 51600000);              // 50000
  float*    exbuf    = (float*)(ws + 51800000);              // 50000
  unsigned* gmax_key = (unsigned*)(ws + 52000000);           // 64
  float*    den      = (float*)(ws + 52000256);              // 64
  float*    x2       = (float*)(ws + 52000512);              // 64*64

  const int T = 256;
  const int rowTiles = N_NODES / 16;                 // 3125 exactly
  const int edgeWaveBlocks = (N_EDGES * 32) / T;     // 200000
  const int nodeWaveBlocks = (N_NODES * 32 + T - 1) / T;

  // degrees + symmetric norm
  deg_init<<<(N_NODES + T - 1) / T, T, 0, stream>>>(deg);
  deg_accum<<<(N_EDGES + T - 1) / T, T, 0, stream>>>(dst, eattr, deg);
  dinv_k<<<(N_NODES + T - 1) / T, T, 0, stream>>>(deg, dinv);

  // ---- layer 1: xw1 = x@W1 ; agg ; relu(+b1) ----
  gemm_wmma<128><<<rowTiles, 256, 0, stream>>>(x, W1, bufA);
  agg_init<128><<<(N_NODES * 128 + T - 1) / T, T, 0, stream>>>(bufA, dinv, bufB);
  edge_scatter<4><<<edgeWaveBlocks, T, 0, stream>>>(bufA, src, dst, eattr, dinv, bufB);
  relu_bias<128><<<(N_NODES * 128 + T - 1) / T, T, 0, stream>>>(bufB, b1);

  // ---- layer 2: xw2 = h@W2 ; agg ; relu(+b2) ----
  gemm_wmma<64><<<rowTiles, 128, 0, stream>>>(bufB, W2, bufA);
  agg_init<64><<<(N_NODES * 64 + T - 1) / T, T, 0, stream>>>(bufA, dinv, bufB);
  edge_scatter<2><<<edgeWaveBlocks, T, 0, stream>>>(bufA, src, dst, eattr, dinv, bufB);
  relu_bias<64><<<(N_NODES * 64 + T - 1) / T, T, 0, stream>>>(bufB, b2);

  // ---- global attention pooling ----
  zero_small<<<1, 256, 0, stream>>>(gmax_key, den, x2);
  gate_kernel<<<nodeWaveBlocks, T, 0, stream>>>(bufB, batch, gW1, gb1, gW2, gb2,
                                                gbuf, gmax_key);
  softmax_num<<<(N_NODES + T - 1) / T, T, 0, stream>>>(gbuf, batch, gmax_key,
                                                       exbuf, den);
  attn_pool<<<nodeWaveBlocks, T, 0, stream>>>(bufB, batch, exbuf, den, x2);

  // ---- head ----
  final_fc<<<N_GRAPHS, 128, 0, stream>>>(x2, fcW1, fcb1, fcW2, fcb2, out);

  (void)in_sizes; (void)n_in; (void)out_size; (void)ws_size;
}